// GeneralizedRingAttractorNoGain_69827578298533
// MI455X (gfx1250) — compile-verified
//
#include <hip/hip_runtime.h>

// Problem constants (match reference)
#define BB 64          // batch
#define TT 512         // time steps
#define NN 256         // neurons
#define AD 8           // action dim
#define J1C 0.1f       // J1
#define LEAKC 0.1f     // DT/TAU

typedef __attribute__((ext_vector_type(2))) float v2f;
typedef __attribute__((ext_vector_type(4))) float v4f;
typedef __attribute__((ext_vector_type(8))) float v8f;
typedef __attribute__((ext_vector_type(4))) int   v4i;

// ---- CDNA5 async LDS->global store path (guarded; fallback = sync stores) --
#if __has_builtin(__builtin_amdgcn_global_store_async_from_lds_b128)
#define USE_ASYNC_STORE 1
#else
#define USE_ASYNC_STORE 0
#endif

__device__ __forceinline__ void wait_async0() {
#if USE_ASYNC_STORE
#if __has_builtin(__builtin_amdgcn_s_wait_asynccnt)
  __builtin_amdgcn_s_wait_asynccnt(0);
#else
  asm volatile("s_wait_asynccnt 0x0" ::: "memory");
#endif
#endif
}

#if USE_ASYNC_STORE
typedef __attribute__((address_space(1))) v4i gas_v4i;   // global int4
typedef __attribute__((address_space(3))) v4i las_v4i;   // LDS int4

__device__ __forceinline__ void async_store_b128(float* gdst, const float* lsrc) {
  __builtin_amdgcn_global_store_async_from_lds_b128(
      (gas_v4i*)gdst, (las_v4i*)lsrc, /*offset=*/0, /*cpol=*/0);
}
#endif

// ---------------------------------------------------------------------------
// f32 WMMA: D(16x16,f32) = A(16x4,f32) * B(4x16,f32) + C
// A layout: lane L<16 -> row M=L, VGPR0=K0,VGPR1=K1 ; lane L>=16 -> K2,K3
// B layout: lane L<16 -> col N=L, VGPR0=K0,VGPR1=K1 ; lane L>=16 -> K2,K3
// C layout: lane L -> col N=L&15 ; VGPR v -> row M = v + 8*(L>>4)
// ---------------------------------------------------------------------------
__device__ __forceinline__ v8f wmma_f32_k4(v2f a, v2f b, v8f c) {
  return __builtin_amdgcn_wmma_f32_16x16x4_f32(
      /*neg_a=*/false, a, /*neg_b=*/false, b,
      /*c_mod=*/(short)0, c, /*reuse_a=*/false, /*reuse_b=*/false);
}

// 16-row tile of (W @ r_tile): Wrow = row (i0 + lane&15) of W; rlds is [j][b].
__device__ __forceinline__ v8f tile_matvec(const float* __restrict__ Wrow,
                                           const float* __restrict__ rlds,
                                           int col, int khi) {
  v8f c = {};
#pragma unroll 4
  for (int k = 0; k < NN; k += 4) {
    const int kk = k + 2 * khi;
    v2f a; a.x = Wrow[kk];            a.y = Wrow[kk + 1];
    v2f b; b.x = rlds[kk * 16 + col]; b.y = rlds[(kk + 1) * 16 + col];
    c = wmma_f32_k4(a, b, c);
  }
  return c;
}

// Same, but A is fused on the fly as (J0 + J1*Wo)
__device__ __forceinline__ v8f tile_matvec_base(const float* __restrict__ j0row,
                                                const float* __restrict__ worow,
                                                const float* __restrict__ rlds,
                                                int col, int khi) {
  v8f c = {};
#pragma unroll 4
  for (int k = 0; k < NN; k += 4) {
    const int kk = k + 2 * khi;
    v2f a;
    a.x = fmaf(J1C, worow[kk],     j0row[kk]);
    a.y = fmaf(J1C, worow[kk + 1], j0row[kk + 1]);
    v2f b; b.x = rlds[kk * 16 + col]; b.y = rlds[(kk + 1) * 16 + col];
    c = wmma_f32_k4(a, b, c);
  }
  return c;
}

__device__ __forceinline__ float fast_tanh(float x) {
  // tanh(x) = 1 - 2/(exp(2x)+1): stable at both +/- saturation
  const float e2 = __expf(2.0f * x);
  return 1.0f - 2.0f / (e2 + 1.0f);
}

// One workgroup = one 16-batch tile; 16 waves, each owns one 16-row M-tile.
// The whole T-step recurrence runs inside the workgroup (no grid sync).
__global__ __launch_bounds__(512, 1)
void GeneralizedRingAttractor_kernel(const float* __restrict__ act,     // [B,T,AD]
                                     const float* __restrict__ r_init,  // [B,N]
                                     const float* __restrict__ Wo,      // [N,N]
                                     const float* __restrict__ Wa,      // [AD,N,N]
                                     const float* __restrict__ J0,      // [N,N]
                                     const float* __restrict__ Wd7,     // [N,N]
                                     float* __restrict__ r_hist,        // [B,T,N]
                                     float* __restrict__ bump_hist)     // [B,T,N]
{
  __shared__ float r_lds[2][NN * 16];   // recurrent state, [j][b], ping-pong
  __shared__ float stage[16 * NN];      // output staging, [b][i] (globally contiguous)
  __shared__ float a_lds[16 * AD];      // A_t for this batch tile
  __shared__ float sumsq[16];           // per-batch ||d7||^2

  const int tid  = threadIdx.x;
  const int lane = tid & 31;
  const int wave = tid >> 5;         // 0..15 -> row tile i0 = wave*16
  const int col  = lane & 15;        // batch column (C/B operand) == A row lane
  const int khi  = lane >> 4;        // K-half select for A/B operands
  const int b0   = blockIdx.x * 16;  // first batch of this tile
  const int i0   = wave * 16;
  const int rowg = i0 + col;         // W row this lane feeds into the A operand

  // Stage r_init -> r_lds[0] (coalesced global read, [j][b] LDS layout)
  for (int e = tid; e < NN * 16; e += 512) {
    const int bb = e >> 8, i = e & (NN - 1);
    r_lds[0][i * 16 + bb] = r_init[(size_t)(b0 + bb) * NN + i];
  }
  __syncthreads();

  int cur = 0;
  for (int t = 0; t < TT; ++t) {
    const int nxt = cur ^ 1;

    // Per-step scalars: A_t tile and norm accumulator
    if (tid < 16 * AD) {
      const int bb = tid >> 3, a = tid & 7;
      a_lds[tid] = act[((size_t)(b0 + bb) * TT + t) * AD + a];
    }
    if (tid < 16) sumsq[tid] = 0.0f;
    __syncthreads();

    const float* rc = r_lds[cur];

    // rec = (J0 + J1*Wo) @ r  +  sum_a A[b,a] * (Wa[a] @ r)
    v8f acc = tile_matvec_base(J0 + (size_t)rowg * NN,
                               Wo + (size_t)rowg * NN, rc, col, khi);
#pragma unroll
    for (int a = 0; a < AD; ++a) {
      const float* warow = Wa + ((size_t)a * NN + rowg) * NN;
      __builtin_prefetch(Wa + ((size_t)((a + 1) & (AD - 1)) * NN + rowg) * NN, 0, 1);
      const v8f y = tile_matvec(warow, rc, col, khi);
      const float s = a_lds[col * AD + a];   // per-column (=batch) scalar
      acc += y * s;
    }

    // Previous step's bump async stores still read `stage`; reclaim it.
    wait_async0();
    __syncthreads();

    // Leaky-rate update; write r_new to B-operand layout AND output staging
#pragma unroll
    for (int v = 0; v < 8; ++v) {
      const int i = i0 + v + 8 * khi;
      const float ro = rc[i * 16 + col];
      const float rn = ro + LEAKC * (fast_tanh(acc[v]) - ro);
      r_lds[nxt][i * 16 + col] = rn;
      stage[col * NN + i]      = rn;
    }
    __syncthreads();

    // r_hist store: async LDS->global (drains while d7 GEMM runs)
#if USE_ASYNC_STORE
#pragma unroll
    for (int c = tid; c < (16 * NN) / 4; c += 512) {   // 16B chunks
      const int bb = c >> 6, o = (c & 63) * 4;
      async_store_b128(r_hist + ((size_t)(b0 + bb) * TT + t) * NN + o,
                       stage + 4 * c);
    }
#else
    for (int e = tid; e < NN * 16; e += 512) {
      const int bb = e >> 8, i = e & (NN - 1);
      r_hist[((size_t)(b0 + bb) * TT + t) * NN + i] = stage[e];
    }
#endif

    // d7 = W_delta7 @ r_new ; accumulate per-batch sum of squares
    const v8f d = tile_matvec(Wd7 + (size_t)rowg * NN, r_lds[nxt], col, khi);

    // r_hist async stores still read `stage`; reclaim before writing d7.
    wait_async0();
    __syncthreads();

    {
      float ss = 0.0f;
#pragma unroll
      for (int v = 0; v < 8; ++v) {
        const int i = i0 + v + 8 * khi;
        stage[col * NN + i] = d[v];
        ss = fmaf(d[v], d[v], ss);
      }
      atomicAdd(&sumsq[col], ss);   // ds_add_f32
    }
    __syncthreads();

    // Normalize in place (vectorized), then fire async bump stores that
    // overlap with the whole next-step recurrent GEMM.
    for (int c = tid; c < (16 * NN) / 4; c += 512) {
      const int bb = c >> 6;
      v4f* sp = (v4f*)stage + c;
      *sp = *sp * rsqrtf(sumsq[bb]);
    }
    __syncthreads();

#if USE_ASYNC_STORE
#pragma unroll
    for (int c = tid; c < (16 * NN) / 4; c += 512) {
      const int bb = c >> 6, o = (c & 63) * 4;
      async_store_b128(bump_hist + ((size_t)(b0 + bb) * TT + t) * NN + o,
                       stage + 4 * c);
    }
#else
    for (int e = tid; e < NN * 16; e += 512) {
      const int bb = e >> 8, i = e & (NN - 1);
      bump_hist[((size_t)(b0 + bb) * TT + t) * NN + i] = stage[e];
    }
    __syncthreads();
#endif

    cur = nxt;
  }
  wait_async0();   // drain before wave exit
}

extern "C" void kernel_launch(void* const* d_in, const int* in_sizes, int n_in,
                              void* d_out, int out_size, void* d_ws, size_t ws_size,
                              hipStream_t stream) {
  (void)in_sizes; (void)n_in; (void)out_size; (void)d_ws; (void)ws_size;
  // setup_inputs() order: action_signal, r_init, Wo, Wa, J0, W_delta7
  const float* act  = (const float*)d_in[0];
  const float* r0   = (const float*)d_in[1];
  const float* Wo   = (const float*)d_in[2];
  const float* Wa   = (const float*)d_in[3];
  const float* J0   = (const float*)d_in[4];
  const float* Wd7  = (const float*)d_in[5];
  float* r_hist    = (float*)d_out;                       // [B,T,N]
  float* bump_hist = r_hist + (size_t)BB * TT * NN;       // [B,T,N]

  dim3 grid(BB / 16);   // 4 batch tiles, each runs the full scan
  dim3 block(512);      // 16 wave32 -> one 16-row M-tile per wave
  hipLaunchKernelGGL(GeneralizedRingAttractor_kernel, grid, block, 0, stream,
                     act, r0, Wo, Wa, J0, Wd7, r_hist, bump_hist);
}